// EquiTritonModel_67508295958936
// MI455X (gfx1250) — compile-verified
//
#include <hip/hip_runtime.h>
#include <math.h>

// ---- model constants -------------------------------------------------------
#define CUT_R   6.0f
#define PI_F    3.14159265358979323846f
#define PIOVC   (PI_F / CUT_R)           // pi/cut, folded to a multiply
#define INV4PI  0.28209479177387814f     // 1/sqrt(4*pi)
#define SQ3I4P  0.48860251190291992f     // sqrt(3)/sqrt(4*pi)
#define RSQ32   0.17677669529663687f     // 1/sqrt(32)  (n0 == n1)

typedef float v2f __attribute__((ext_vector_type(2)));
typedef float v8f __attribute__((ext_vector_type(8)));

// D = A(16x4,f32) * B(4x16,f32) + C(16x16,f32), full-rate f32 WMMA on gfx1250.
__device__ __forceinline__ v8f wmma_f32(v2f a, v2f b, v8f c) {
  return __builtin_amdgcn_wmma_f32_16x16x4_f32(false, a, false, b,
                                               (short)0, c, false, false);
}

// single v_rcp_f32 instead of the IEEE v_div_fixup expansion
__device__ __forceinline__ float fast_rcp(float x) {
  return __builtin_amdgcn_rcpf(x);
}

__device__ __forceinline__ float silu(float x) {
  return x * fast_rcp(1.0f + __expf(-x));
}

// In-wave LDS ordering: all our LDS producer/consumer traffic is within one
// wave, so a DScnt drain (CDNA5 split wait counter) is sufficient -- no
// workgroup barrier needed.
__device__ __forceinline__ void lds_fence() {
  asm volatile("s_wait_dscnt 0" ::: "memory");
}

// ---------------------------------------------------------------------------
__global__ __launch_bounds__(256) void k_zero(float* ws, size_t n, float* out) {
  size_t i = (size_t)blockIdx.x * blockDim.x + threadIdx.x;
  if (i == 0) out[0] = 0.0f;
  const size_t stride = (size_t)gridDim.x * blockDim.x;
  for (; i < n; i += stride) ws[i] = 0.0f;
}

// ---------------------------------------------------------------------------
// Pass 1: per-edge  basis -> silu(basis@W1/4) -> h@W2/4 -> contract with xs
// over D -> atomic scatter into zs0 (N x 16) and zs1 (N x 16 x 3).
// One wave handles 16 edges (= the WMMA M dimension).
__global__ __launch_bounds__(256) void k_pass1(
    const int*   __restrict__ an,     const float* __restrict__ coords,
    const int*   __restrict__ ei,     const float* __restrict__ emb,
    const float* __restrict__ w1,     const float* __restrict__ w2,
    float* __restrict__ zs0, float* __restrict__ zs1, int E)
{
  __shared__ float sh[8 * 832];
  const int lane = threadIdx.x & 31;
  const int wv   = threadIdx.x >> 5;
  const int m    = lane & 15;          // edge row in tile == B/C/D column
  const int half = lane >> 4;

  float* hA  = sh + wv * 832;          // 16x16 : layer-1 result, A-layout stage
  float* xsT = hA  + 256;              // 16x32 : xs rows for this edge tile
  float* unT = xsT + 512;              // 16x3  : unit vectors
  int*   dsT = (int*)(unT + 48);       // 16    : dst node ids

  // warm L1/L2 with the big weight matrix (64 KB, fully L2 resident)
  __builtin_prefetch(w2 + (size_t)threadIdx.x * 64, 0, 0);

  int e = (blockIdx.x * 8 + wv) * 16 + m;
  const bool valid = (e < E);
  if (!valid) e = E - 1;
  const int s_ = ei[e];
  const int d_ = ei[E + e];
  const float vx = coords[s_ * 3 + 0] - coords[d_ * 3 + 0];
  const float vy = coords[s_ * 3 + 1] - coords[d_ * 3 + 1];
  const float vz = coords[s_ * 3 + 2] - coords[d_ * 3 + 2];
  const float dist = sqrtf(vx * vx + vy * vy + vz * vz);
  const float rinv = fast_rcp(fmaxf(dist, 1e-9f));
  const float ux = vx * rinv, uy = vy * rinv, uz = vz * rinv;
  // bessel prefactor: sqrt(2/cut)*sqrt(NB)/safe, cutoff mask, invalid-edge mask
  const float bco = (valid && dist < CUT_R)
                        ? (sqrtf(2.0f / CUT_R) * 4.0f * rinv) : 0.0f;
  const float th = dist * PIOVC;       // sin argument step (multiply, no divide)

  if (half == 0) {
    unT[m * 3 + 0] = ux; unT[m * 3 + 1] = uy; unT[m * 3 + 2] = uz;
    dsT[m] = d_;
  }
  {  // both halves cooperate: 16 columns of xs each
    const int a = an[s_];
    #pragma unroll
    for (int t = 0; t < 16; ++t)
      xsT[m * 32 + half * 16 + t] = emb[a * 32 + half * 16 + t];
  }

  // A-matrix chunks (16x4 f32 layout: K = 4s + 2*half + j) = bessel basis
  v2f aCh[4];
  #pragma unroll
  for (int s = 0; s < 4; ++s)
    #pragma unroll
    for (int j = 0; j < 2; ++j) {
      const int k = s * 4 + 2 * half + j;
      aCh[s][j] = bco * __sinf((float)(k + 1) * th);
    }

  // layer 1: pre = basis @ fc0_w1  (K=16 via 4 chained K=4 WMMAs)
  v8f pre = {0, 0, 0, 0, 0, 0, 0, 0};
  #pragma unroll
  for (int s = 0; s < 4; ++s) {
    const int k0 = s * 4 + 2 * half;
    v2f b; b[0] = w1[(k0 + 0) * 16 + m]; b[1] = w1[(k0 + 1) * 16 + m];
    pre = wmma_f32(aCh[s], b, pre);
  }
  // h = silu(pre/4); transpose C/D layout -> A layout through LDS
  #pragma unroll
  for (int r = 0; r < 8; ++r)
    hA[(r + 8 * half) * 16 + m] = silu(pre[r] * 0.25f);
  lds_fence();

  v2f hCh[4];
  #pragma unroll
  for (int s = 0; s < 4; ++s) {
    hCh[s][0] = hA[m * 16 + s * 4 + 2 * half + 0];
    hCh[s][1] = hA[m * 16 + s * 4 + 2 * half + 1];
  }

  // layer 2: w = h @ fc0_w2 (16x1024); column-tile c==d gives wa/wb slabs,
  // contracted immediately against xs[:,c] (row-broadcast from LDS).
  v8f macc0 = {0, 0, 0, 0, 0, 0, 0, 0};
  v8f macc1 = {0, 0, 0, 0, 0, 0, 0, 0};
  for (int c = 0; c < 32; ++c) {
    v8f ta = {0, 0, 0, 0, 0, 0, 0, 0};
    v8f tb = {0, 0, 0, 0, 0, 0, 0, 0};
    #pragma unroll
    for (int s = 0; s < 4; ++s) {
      const int k0 = s * 4 + 2 * half;
      const float* p = w2 + k0 * 1024 + c * 16 + m;
      v2f ba; ba[0] = p[0];   ba[1] = p[1024];        // wa columns
      v2f bb; bb[0] = p[512]; bb[1] = p[512 + 1024];  // wb columns
      ta = wmma_f32(hCh[s], ba, ta);
      tb = wmma_f32(hCh[s], bb, tb);
    }
    #pragma unroll
    for (int r = 0; r < 8; ++r) {
      const float xv = xsT[(r + 8 * half) * 32 + c];
      macc0[r] += xv * ta[r];
      macc1[r] += xv * tb[r];
    }
  }

  // m0 = (xs.wa)*y0*n0, m1 = (xs.wb)*y1*n0 ; z = segsum/DEG
  const float S0 = 0.25f * RSQ32 * INV4PI * 0.25f;  // fc * n0 * y0 * 1/DEG
  const float S1 = 0.25f * RSQ32 * 0.25f * SQ3I4P;  // fc * n0 * 1/DEG * sqrt3*y0
  #pragma unroll
  for (int r = 0; r < 8; ++r) {
    const int row = r + 8 * half;
    const int dd  = dsT[row];
    const float u0 = unT[row * 3 + 0];
    const float u1 = unT[row * 3 + 1];
    const float u2 = unT[row * 3 + 2];
    atomicAdd(&zs0[dd * 16 + m], macc0[r] * S0);
    const float mb = macc1[r] * S1;
    atomicAdd(&zs1[(dd * 16 + m) * 3 + 0], mb * u0);
    atomicAdd(&zs1[(dd * 16 + m) * 3 + 1], mb * u1);
    atomicAdd(&zs1[(dd * 16 + m) * 3 + 2], mb * u2);
  }
}

// ---------------------------------------------------------------------------
// Pass 2: final z0 only needs o0 = (s@w1_ *y0 + dot@w4_)*n1 -> only columns
// [0,256) and [768,1024) of fc1_w2 (o1/z1-final are dead in the readout).
__global__ __launch_bounds__(256) void k_pass2(
    const float* __restrict__ coords, const int* __restrict__ ei,
    const float* __restrict__ w1,     const float* __restrict__ w2,
    const float* __restrict__ zs0,    const float* __restrict__ zs1,
    float* __restrict__ zacc0, int E)
{
  __shared__ float sh[8 * 784];
  const int lane = threadIdx.x & 31;
  const int wv   = threadIdx.x >> 5;
  const int m    = lane & 15;
  const int half = lane >> 4;

  float* hA  = sh + wv * 784;       // 16x16 layer-1 stage
  float* sT  = hA + 256;            // 16x16 s  = z0[src]
  float* dT  = sT + 256;            // 16x16 dot = INV4PI * v.unit
  int*   dsT = (int*)(dT + 256);    // 16 dst ids

  __builtin_prefetch(w2 + (size_t)threadIdx.x * 64, 0, 0);

  int e = (blockIdx.x * 8 + wv) * 16 + m;
  const bool valid = (e < E);
  if (!valid) e = E - 1;
  const int s_ = ei[e];
  const int d_ = ei[E + e];
  const float vx = coords[s_ * 3 + 0] - coords[d_ * 3 + 0];
  const float vy = coords[s_ * 3 + 1] - coords[d_ * 3 + 1];
  const float vz = coords[s_ * 3 + 2] - coords[d_ * 3 + 2];
  const float dist = sqrtf(vx * vx + vy * vy + vz * vz);
  const float rinv = fast_rcp(fmaxf(dist, 1e-9f));
  const float ux = vx * rinv, uy = vy * rinv, uz = vz * rinv;
  const float bco = (valid && dist < CUT_R)
                        ? (sqrtf(2.0f / CUT_R) * 4.0f * rinv) : 0.0f;
  const float th = dist * PIOVC;

  if (half == 0) dsT[m] = d_;
  #pragma unroll
  for (int t = 0; t < 8; ++t) {     // halves split the 16 h-channels
    const int h = half * 8 + t;
    sT[m * 16 + h] = zs0[s_ * 16 + h];
    const float q0 = zs1[(s_ * 16 + h) * 3 + 0];
    const float q1 = zs1[(s_ * 16 + h) * 3 + 1];
    const float q2 = zs1[(s_ * 16 + h) * 3 + 2];
    dT[m * 16 + h] = INV4PI * (q0 * ux + q1 * uy + q2 * uz);
  }

  v2f aCh[4];
  #pragma unroll
  for (int s = 0; s < 4; ++s)
    #pragma unroll
    for (int j = 0; j < 2; ++j) {
      const int k = s * 4 + 2 * half + j;
      aCh[s][j] = bco * __sinf((float)(k + 1) * th);
    }

  v8f pre = {0, 0, 0, 0, 0, 0, 0, 0};
  #pragma unroll
  for (int s = 0; s < 4; ++s) {
    const int k0 = s * 4 + 2 * half;
    v2f b; b[0] = w1[(k0 + 0) * 16 + m]; b[1] = w1[(k0 + 1) * 16 + m];
    pre = wmma_f32(aCh[s], b, pre);
  }
  #pragma unroll
  for (int r = 0; r < 8; ++r)
    hA[(r + 8 * half) * 16 + m] = silu(pre[r] * 0.25f);
  lds_fence();

  v2f hCh[4];
  #pragma unroll
  for (int s = 0; s < 4; ++s) {
    hCh[s][0] = hA[m * 16 + s * 4 + 2 * half + 0];
    hCh[s][1] = hA[m * 16 + s * 4 + 2 * half + 1];
  }

  v8f accS = {0, 0, 0, 0, 0, 0, 0, 0};
  v8f accD = {0, 0, 0, 0, 0, 0, 0, 0};
  for (int c = 0; c < 16; ++c) {
    v8f tS = {0, 0, 0, 0, 0, 0, 0, 0};
    v8f tD = {0, 0, 0, 0, 0, 0, 0, 0};
    #pragma unroll
    for (int s = 0; s < 4; ++s) {
      const int k0 = s * 4 + 2 * half;
      const float* p = w2 + k0 * 1024 + c * 16 + m;
      v2f b1; b1[0] = p[0];   b1[1] = p[1024];        // w1_ tile (h=c)
      v2f b4; b4[0] = p[768]; b4[1] = p[768 + 1024];  // w4_ tile (h=c)
      tS = wmma_f32(hCh[s], b1, tS);
      tD = wmma_f32(hCh[s], b4, tD);
    }
    #pragma unroll
    for (int r = 0; r < 8; ++r) {
      accS[r] += sT[(r + 8 * half) * 16 + c] * tS[r];
      accD[r] += dT[(r + 8 * half) * 16 + c] * tD[r];
    }
  }

  const float S2 = 0.25f * RSQ32 * 0.25f;  // fc scale * n1 * 1/DEG
  #pragma unroll
  for (int r = 0; r < 8; ++r) {
    const int dd = dsT[r + 8 * half];
    atomicAdd(&zacc0[dd * 16 + m], (INV4PI * accS[r] + accD[r]) * S2);
  }
}

// ---------------------------------------------------------------------------
__global__ __launch_bounds__(256) void k_readout(
    const float* __restrict__ zs0, const float* __restrict__ zacc0,
    const float* __restrict__ wr, float* __restrict__ out, int N)
{
  const int n = blockIdx.x * blockDim.x + threadIdx.x;
  float acc = 0.0f;
  if (n < N) {
    #pragma unroll
    for (int h = 0; h < 16; ++h)
      acc += (zs0[n * 16 + h] + zacc0[n * 16 + h]) * wr[h];
    acc *= 0.25f;  // 1/sqrt(H)
  }
  #pragma unroll
  for (int off = 16; off > 0; off >>= 1) acc += __shfl_down(acc, off, 32);
  if ((threadIdx.x & 31) == 0) atomicAdd(out, acc);
}

// ---------------------------------------------------------------------------
extern "C" void kernel_launch(void* const* d_in, const int* in_sizes, int n_in,
                              void* d_out, int out_size, void* d_ws, size_t ws_size,
                              hipStream_t stream) {
  const int*   an     = (const int*)  d_in[0];
  const float* coords = (const float*)d_in[1];
  const int*   ei     = (const int*)  d_in[2];
  const float* emb    = (const float*)d_in[3];
  const float* f0w1   = (const float*)d_in[4];
  const float* f0w2   = (const float*)d_in[5];
  const float* f1w1   = (const float*)d_in[6];
  const float* f1w2   = (const float*)d_in[7];
  const float* wr     = (const float*)d_in[8];
  float* out = (float*)d_out;

  const int N = in_sizes[0];
  const int E = in_sizes[2] / 2;

  float* zs0   = (float*)d_ws;                 // N*16
  float* zs1   = zs0 + (size_t)N * 16;         // N*48
  float* zacc0 = zs1 + (size_t)N * 48;         // N*16
  const size_t zn = (size_t)N * 80;

  k_zero<<<512, 256, 0, stream>>>((float*)d_ws, zn, out);

  const int eb = (E + 127) / 128;              // 8 waves * 16 edges per block
  k_pass1<<<eb, 256, 0, stream>>>(an, coords, ei, emb, f0w1, f0w2, zs0, zs1, E);
  k_pass2<<<eb, 256, 0, stream>>>(coords, ei, f1w1, f1w2, zs0, zs1, zacc0, E);
  k_readout<<<(N + 255) / 256, 256, 0, stream>>>(zs0, zacc0, wr, out, N);
}